// EntityEncoder_18700287607031
// MI455X (gfx1250) — compile-verified
//
#include <hip/hip_runtime.h>
#include <hip/hip_bf16.h>

#define BS  32
#define J   4
#define L   512
#define D   768
#define K   4
#define OUT 256
#define M_ROWS (BS * J * K)   // 512 entity rows

typedef __attribute__((ext_vector_type(2))) float v2f;
typedef __attribute__((ext_vector_type(8))) float v8f;

// ---------------------------------------------------------------------------
// Kernel 1: ragged span mean-pool.
// One block per entity row (b*J*K + j*K + k); 192 threads = one float4 per
// thread covers D=768 (192*4). Coalesced: 192 lanes * 16B = 3072B contiguous
// per token row. This is the HBM-bound stage (~50 MB of span reads).
// ---------------------------------------------------------------------------
__global__ __launch_bounds__(192)
void entity_pool_kernel(const float* __restrict__ z,     // [BS,J,L,D]
                        const int*   __restrict__ sep,   // [BS,J,K]
                        float*       __restrict__ pooled)// [M_ROWS, D]
{
    const int row = blockIdx.x;          // 0..511  == (b*J + j)*K + k
    const int k   = row & (K - 1);
    const int bj  = row >> 2;            // b*J + j, 0..127

    const int start = (k == 0) ? 1 : (sep[bj * K + k - 1] + 1);
    const int end   = sep[bj * K + k];   // exclusive

    const int tid = threadIdx.x;         // 0..191
    const float* base = z + (size_t)bj * L * D + tid * 4;

    float4 acc = make_float4(0.f, 0.f, 0.f, 0.f);
    for (int l = start; l < end; ++l) {
        const float4 v = *(const float4*)(base + (size_t)l * D);
        acc.x += v.x; acc.y += v.y; acc.z += v.z; acc.w += v.w;
    }
    const float inv = 1.0f / (float)(end - start);
    acc.x *= inv; acc.y *= inv; acc.z *= inv; acc.w *= inv;

    *(float4*)(pooled + (size_t)row * D + tid * 4) = acc;
}

// ---------------------------------------------------------------------------
// Kernel 2: pooled[512,768] @ W[768,256] + b via V_WMMA_F32_16X16X4_F32.
// One wave32 per 16x16 output tile; 32x16 tiles total = 512 waves.
// A fragment: lane holds A[m = lane&15][k0 + (lane>>4)*2 + {0,1}] (float2 load).
// B fragment: lane holds B[k0 + (lane>>4)*2 + {0,1}][n = lane&15].
// C/D: vgpr r -> row = tileM + r + (lane>=16 ? 8 : 0), col = tileN + (lane&15).
// ---------------------------------------------------------------------------
__global__ __launch_bounds__(256)
void entity_gemm_wmma_kernel(const float* __restrict__ A,    // [M_ROWS, D]
                             const float* __restrict__ W,    // [D, OUT]
                             const float* __restrict__ bias, // [OUT]
                             float*       __restrict__ out)  // [M_ROWS, OUT]
{
    const int lane = threadIdx.x & 31;
    const int wave = blockIdx.x * (blockDim.x >> 5) + (threadIdx.x >> 5); // 0..511
    const int tm   = (wave >> 4) << 4;   // tile row base: 0..496
    const int tn   = (wave & 15) << 4;   // tile col base: 0..240

    const int l15  = lane & 15;
    const int hi   = lane >> 4;          // 0 or 1
    const int mrow = tm + l15;
    const int ncol = tn + l15;

    const float* arow = A + (size_t)mrow * D + (hi << 1);
    const float* wcol = W + (size_t)(hi << 1) * OUT + ncol;

    v8f c = {0.f, 0.f, 0.f, 0.f, 0.f, 0.f, 0.f, 0.f};

#pragma unroll 8
    for (int k0 = 0; k0 < D; k0 += 4) {
        v2f a = *(const v2f*)(arow + k0);
        v2f b;
        b.x = wcol[(size_t)k0 * OUT];
        b.y = wcol[(size_t)k0 * OUT + OUT];
        // D = A(16x4) * B(4x16) + C ; fp32 WMMA path (no downcast needed)
        c = __builtin_amdgcn_wmma_f32_16x16x4_f32(
                /*neg_a=*/false, a, /*neg_b=*/false, b,
                /*c_mod=*/(short)0, c, /*reuse_a=*/false, /*reuse_b=*/false);
    }

    const float bv = bias[ncol];
#pragma unroll
    for (int r = 0; r < 8; ++r) {
        const int row = tm + r + (hi << 3);
        out[(size_t)row * OUT + ncol] = c[r] + bv;
    }
}

// ---------------------------------------------------------------------------
// Launch: pooling -> ws, then WMMA GEMM -> d_out (same stream, ordered).
// ws usage: 512*768*4 = 1.5 MB of pooled fp32.
// ---------------------------------------------------------------------------
extern "C" void kernel_launch(void* const* d_in, const int* in_sizes, int n_in,
                              void* d_out, int out_size, void* d_ws, size_t ws_size,
                              hipStream_t stream) {
    const float* z    = (const float*)d_in[0];   // zipped_entity [32,4,512,768]
    const int*   sep  = (const int*)  d_in[1];   // entity_token_sep_idx [32,4,4]
    const float* W    = (const float*)d_in[2];   // [768,256]
    const float* bias = (const float*)d_in[3];   // [256]
    float*       out  = (float*)d_out;           // [512,256]
    float*       pooled = (float*)d_ws;          // [512,768]

    entity_pool_kernel<<<M_ROWS, 192, 0, stream>>>(z, sep, pooled);
    entity_gemm_wmma_kernel<<<M_ROWS / 8, 256, 0, stream>>>(pooled, W, bias, out);
}